// AdaptivePruner_36558761624188
// MI455X (gfx1250) — compile-verified
//
#include <hip/hip_runtime.h>
#include <cmath>

// ---------------------------------------------------------------------------
// AdaptivePruner for MI455X (gfx1250, wave32, WMMA).
// Memory-bound (~604 MB traffic @ 23.3 TB/s ~ 26us); fused single pass:
//  - level 1: y1 = 8-tap stride-2 conv of patches
//  - level 2: y2 = composite 22-tap stride-4 conv (exact fusion of 2 levels)
// Each wave computes one 16(outputs) x 16(channels) tile of the result with
// V_WMMA_F32_16X16X4_F32: banded filter matrix (A) x input rows (B).
// A-fragments are tile-invariant -> precomputed per-block LDS table.
// Interior tiles (wave-uniform test) use a single base pointer + immediate
// offsets for all row loads; only edge tiles pay clamp+select costs.
// ---------------------------------------------------------------------------

typedef float v2f __attribute__((ext_vector_type(2)));
typedef float v8f __attribute__((ext_vector_type(8)));

#define BATCH 32
#define NPAT  4096
#define DIM   768
#define L1    2051          // (4096+7)/2
#define L2    1029          // (2051+7)/2
#define NOUT  2052          // 1 cls + L1

// KERNEL_LO = DEC_LO reversed (cross-correlation kernel, no flip in lax.conv)
__constant__ float KLO[8] = {
     0.2303778133088965f,   0.7148465705529157f,
     0.6308807679298589f,  -0.027983769416859854f,
    -0.18703481171909309f,  0.030841381835560764f,
     0.0328830116668852f,  -0.010597401785069032f };

__device__ __forceinline__ int iclamp(int v, int lo, int hi) {
    return v < lo ? lo : (v > hi ? hi : v);
}

__device__ __forceinline__ v8f wmma4(v2f a, v2f b, v8f c) {
    return __builtin_amdgcn_wmma_f32_16x16x4_f32(
        false, a, false, b, (short)0, c, false, false);
}

// --------------------------- entropy per batch -----------------------------
__global__ void ent_kernel(const float* __restrict__ attn, float* __restrict__ ent) {
    __shared__ float red[256];
    const int b = blockIdx.x;
    const float* row = attn + (size_t)b * NPAT;
    float s = 0.f;
    for (int i = threadIdx.x; i < NPAT; i += 256) {
        float a = row[i];
        s += a * log2f(a + 1e-9f);
    }
    red[threadIdx.x] = s;
    __syncthreads();
    for (int st = 128; st > 0; st >>= 1) {
        if (threadIdx.x < st) red[threadIdx.x] += red[threadIdx.x + st];
        __syncthreads();
    }
    if (threadIdx.x == 0) ent[b] = -red[0];
}

// --------------------- mean/std(ddof=1) -> level per batch -----------------
__global__ void lvl_kernel(const float* __restrict__ ent, int* __restrict__ lvl) {
    if (threadIdx.x != 0) return;
    float m = 0.f;
    for (int i = 0; i < BATCH; ++i) m += ent[i];
    m *= (1.0f / BATCH);
    float v = 0.f;
    for (int i = 0; i < BATCH; ++i) { float d = ent[i] - m; v += d * d; }
    v *= (1.0f / (BATCH - 1));
    float s = sqrtf(v);
    float thr0 = m - 1.5f * s;
    float thr1 = m - 0.5f * s;
    for (int i = 0; i < BATCH; ++i) {
        int l = (ent[i] < thr0 ? 1 : 0) + (ent[i] < thr1 ? 1 : 0);
        if (s < 1e-6f) l = 1;
        if (l < 1) l = 1;
        if (l > 2) l = 2;
        lvl[i] = l;
    }
}

// ------------------------------ main DWT -----------------------------------
// grid = (6, 129, 32), block = 256 (8 waves). Wave w owns d-tile blockIdx.x*8+w.
__global__ __launch_bounds__(256)
void dwt_wmma_kernel(const float* __restrict__ x, const int* __restrict__ lvl,
                     float* __restrict__ out) {
    __shared__ float s_w[8];        // level-1 taps
    __shared__ float s_c2[24];      // composite 22 taps (padded)
    __shared__ v2f   s_a1[10][32];  // level-1 A-fragment table  [chunk][lane]
    __shared__ v2f   s_a2[21][32];  // level-2 A-fragment table  [chunk][lane]

    const int tid = threadIdx.x;

    // phase 1: tap tables
    if (tid < 8) s_w[tid] = KLO[tid];
    if (tid >= 32 && tid < 56) {          // c2[m] = sum_k w[k]*w[m-2k]
        int mm = tid - 32;
        float a = 0.f;
        #pragma unroll
        for (int k = 0; k < 8; ++k) {
            int j = mm - 2 * k;
            if (j >= 0 && j < 8) a += KLO[k] * KLO[j];
        }
        s_c2[mm] = a;                     // m=22,23 -> 0
    }
    __syncthreads();

    // phase 2: per-lane A-fragment tables (tile-invariant, built once)
    for (int e = tid; e < 31 * 32; e += 256) {
        const int chunk = e >> 5, ln = e & 31;
        const int mm = ln & 15, klc = (ln >> 4) << 1;
        v2f a;
        if (chunk < 10) {
            const int t0 = klc - 2 * mm + 4 * chunk;
            a.x = ((unsigned)t0       < 8u) ? s_w[iclamp(t0,     0, 7)] : 0.f;
            a.y = ((unsigned)(t0 + 1) < 8u) ? s_w[iclamp(t0 + 1, 0, 7)] : 0.f;
            s_a1[chunk][ln] = a;
        } else {
            const int c = chunk - 10;
            const int t0 = klc - 4 * mm + 4 * c;
            a.x = ((unsigned)t0       < 22u) ? s_c2[iclamp(t0,     0, 21)] : 0.f;
            a.y = ((unsigned)(t0 + 1) < 22u) ? s_c2[iclamp(t0 + 1, 0, 21)] : 0.f;
            s_a2[c][ln] = a;
        }
    }
    __syncthreads();

    const int b    = blockIdx.z;
    const int o0   = blockIdx.y * 16;
    const int wv   = tid >> 5;
    const int lane = tid & 31;
    const int d0   = (blockIdx.x * 8 + wv) * 16;
    const int m    = lane & 15;              // A: M row / B: N col / store: d col
    const int kl   = (lane >> 4) << 1;       // K half: 0 (lanes 0-15) / 2 (16-31)
    const int d    = d0 + m;

    const float* xb = x + ((size_t)b * (NPAT + 1) + 1) * DIM;  // patches base
    const int L = lvl[b];

    v8f acc = {0.f, 0.f, 0.f, 0.f, 0.f, 0.f, 0.f, 0.f};

    if (L == 1) {
        const int n0 = 2 * o0 - 6;          // window rows [n0, n0+40)
        if (n0 >= 0 && n0 + 40 <= NPAT) {
            // interior: single base pointer, all loads via immediate offsets
            const float* px = xb + (size_t)(n0 + kl) * DIM + d;
            v2f av[10], bv[10];
            #pragma unroll
            for (int c = 0; c < 10; ++c) av[c] = s_a1[c][lane];
            #pragma unroll
            for (int c = 0; c < 10; ++c) {
                bv[c].x = px[c * 4 * DIM];
                bv[c].y = px[c * 4 * DIM + DIM];
            }
            #pragma unroll
            for (int c = 0; c < 10; ++c) acc = wmma4(av[c], bv[c], acc);
        } else {
            // edge: clamp address, select value to 0 when out of range
            #pragma unroll
            for (int c = 0; c < 10; ++c) {
                const v2f a = s_a1[c][lane];
                const int n = n0 + kl + 4 * c;
                const float v0 = xb[(size_t)iclamp(n,     0, NPAT - 1) * DIM + d];
                const float v1 = xb[(size_t)iclamp(n + 1, 0, NPAT - 1) * DIM + d];
                v2f bv;
                bv.x = ((unsigned)n       < (unsigned)NPAT) ? v0 : 0.f;
                bv.y = ((unsigned)(n + 1) < (unsigned)NPAT) ? v1 : 0.f;
                acc = wmma4(a, bv, acc);
            }
        }
    } else {
        const int n0 = 4 * o0 - 18;         // window rows [n0, n0+84)
        if (n0 >= NPAT) {
            // entire window past the input: pure y2p zero padding, acc stays 0
        } else if (n0 >= 0 && n0 + 84 <= NPAT) {
            const float* px = xb + (size_t)(n0 + kl) * DIM + d;
            v2f av[21], bv[21];
            #pragma unroll
            for (int c = 0; c < 21; ++c) av[c] = s_a2[c][lane];
            #pragma unroll
            for (int c = 0; c < 21; ++c) {
                bv[c].x = px[c * 4 * DIM];
                bv[c].y = px[c * 4 * DIM + DIM];
            }
            #pragma unroll
            for (int c = 0; c < 21; ++c) acc = wmma4(av[c], bv[c], acc);
        } else {
            #pragma unroll
            for (int c = 0; c < 21; ++c) {
                const v2f a = s_a2[c][lane];
                const int n = n0 + kl + 4 * c;
                const float v0 = xb[(size_t)iclamp(n,     0, NPAT - 1) * DIM + d];
                const float v1 = xb[(size_t)iclamp(n + 1, 0, NPAT - 1) * DIM + d];
                v2f bv;
                bv.x = ((unsigned)n       < (unsigned)NPAT) ? v0 : 0.f;
                bv.y = ((unsigned)(n + 1) < (unsigned)NPAT) ? v1 : 0.f;
                acc = wmma4(a, bv, acc);
            }
        }
    }

    // D layout: VGPR r -> M = r (lanes 0-15) or r+8 (lanes 16-31), N = lane&15
    float* ob = out + (size_t)b * NOUT * DIM + (size_t)(o0 + 1) * DIM + d;
    const int ro = (lane < 16) ? 0 : 8;
    #pragma unroll
    for (int r = 0; r < 8; ++r) {
        const int o = o0 + ro + r;
        if (o < L1) ob[(size_t)(ro + r) * DIM] = acc[r];
    }
}

// --------------------- cls-token copy + mask fill --------------------------
__global__ void cls_mask_kernel(const float* __restrict__ x, const int* __restrict__ lvl,
                                float* __restrict__ out, float* __restrict__ mask) {
    const int i = blockIdx.x * 256 + threadIdx.x;
    if (i < BATCH * DIM) {
        const int b = i / DIM, dd = i % DIM;
        out[(size_t)b * NOUT * DIM + dd] = x[(size_t)b * (NPAT + 1) * DIM + dd];
    }
    if (i < BATCH * NOUT) {
        const int b = i / NOUT, j = i % NOUT;
        const int len = (lvl[b] == 1) ? L1 : L2;
        mask[i] = (j == 0 || (j - 1) < len) ? 1.0f : 0.0f;
    }
}

extern "C" void kernel_launch(void* const* d_in, const int* in_sizes, int n_in,
                              void* d_out, int out_size, void* d_ws, size_t ws_size,
                              hipStream_t stream) {
    const float* x    = (const float*)d_in[0];   // (32, 4097, 768) f32
    const float* attn = (const float*)d_in[1];   // (32, 4096)      f32
    float* out  = (float*)d_out;                              // final_x flat
    float* mask = out + (size_t)BATCH * NOUT * DIM;           // then mask flat
    float* ent  = (float*)d_ws;                               // 32 floats
    int*   lvl  = (int*)d_ws + BATCH;                         // 32 ints

    ent_kernel<<<BATCH, 256, 0, stream>>>(attn, ent);
    lvl_kernel<<<1, 32, 0, stream>>>(ent, lvl);

    dim3 grid(DIM / (16 * 8), (L1 + 15) / 16, BATCH);         // (6, 129, 32)
    dwt_wmma_kernel<<<grid, 256, 0, stream>>>(x, lvl, out);

    const int nmx = BATCH * NOUT;                             // 65664 covers cls too
    cls_mask_kernel<<<(nmx + 255) / 256, 256, 0, stream>>>(x, lvl, out, mask);
}